// GATv1_34600256537481
// MI455X (gfx1250) — compile-verified
//
#include <hip/hip_runtime.h>

typedef __attribute__((ext_vector_type(2))) float v2f;
typedef __attribute__((ext_vector_type(4))) float v4f;
typedef __attribute__((ext_vector_type(8))) float v8f;
typedef __attribute__((ext_vector_type(4))) int   v4i;

#define NEG_SLOPE 0.2f

#if defined(__HIP_DEVICE_COMPILE__) && __has_builtin(__builtin_amdgcn_global_load_async_to_lds_b128)
#define HAVE_ASYNC_LDS 1
typedef __attribute__((address_space(1))) v4i gv4i_t;   // global int4
typedef __attribute__((address_space(3))) v4i lv4i_t;   // LDS int4
#else
#define HAVE_ASYNC_LDS 0
#endif

__device__ __forceinline__ float leakyf(float v) { return v > 0.f ? v : v * NEG_SLOPE; }

// Float atomic max via ordered-int trick: works for all finite floats.
__device__ __forceinline__ void atomicMaxF(float* addr, float v) {
  if (v >= 0.f) atomicMax((int*)addr, __float_as_int(v));
  else          atomicMin((unsigned int*)addr, __float_as_uint(v));
}

__device__ __forceinline__ void atomAddF(float* addr, float v) {
  __hip_atomic_fetch_add(addr, v, __ATOMIC_RELAXED, __HIP_MEMORY_SCOPE_AGENT);
}

// Async copy of 16 bytes global -> LDS (falls back to a synchronous copy).
__device__ __forceinline__ void copy16_g2l(const float* gsrc, float* ldst) {
#if HAVE_ASYNC_LDS
  __builtin_amdgcn_global_load_async_to_lds_b128(
      (gv4i_t*)(uintptr_t)gsrc,        // inttoptr to AS1 (64-bit, value-preserving)
      (lv4i_t*)ldst,                   // addrspacecast generic -> AS3
      0, 0);
#else
  *(v4f*)ldst = *(const v4f*)gsrc;
#endif
}

__device__ __forceinline__ void wait_async_lds() {
#if HAVE_ASYNC_LDS
  asm volatile("s_wait_asynccnt 0x0" ::: "memory");
#endif
}

// ---------------------------------------------------------------------------
// GEMM1: h1[N,64] = x[N,512] @ W1[512,64]   (fp32 WMMA 16x16x4)
// Block = 256 threads = 8 waves. Block covers 32 rows (2 row-tiles x 4 col-
// tiles, one 16x16 tile per wave). The 32x512 x-tile is staged through LDS in
// double-buffered 32x64 chunks using async global->LDS copies, so the 4 waves
// sharing a row tile read A fragments from LDS instead of 4x-redundant global
// loads. W1 (128 KB, shared by every block) is read from global / L2.
// ---------------------------------------------------------------------------
#define KCHUNK 64
__global__ __launch_bounds__(256) void gemm1_wmma(const float* __restrict__ x,
                                                  const float* __restrict__ W,
                                                  float* __restrict__ h, int n) {
  __shared__ float xtile[2][32 * KCHUNK];   // 2 x 8 KB

  const int tid  = threadIdx.x;
  const int wave = tid >> 5;
  const int lane = tid & 31;
  const int rowTile = blockIdx.x * 2 + (wave >> 2);
  const int colTile = wave & 3;
  const int l15 = lane & 15;
  const int hi  = lane >> 4;                 // 0: K,K+1   1: K+2,K+3
  const int row0 = blockIdx.x * 32;
  const bool waveOK = (rowTile * 16) < n;    // wave-uniform

  // Staging: 32 rows x 64 cols = 512 float4s; each thread copies f4 #tid and
  // #tid+256.  f4 index i -> row = i>>4, col = (i&15)*4  (256B coalesced runs)
  const int sr0 = tid >> 4;
  const int sr1 = sr0 + 16;
  const int sc  = (tid & 15) * 4;
  const int gr0 = (row0 + sr0 < n) ? row0 + sr0 : n - 1;  // clamp OOB rows
  const int gr1 = (row0 + sr1 < n) ? row0 + sr1 : n - 1;

  const float* wc = W + (size_t)(hi * 2) * 64 + colTile * 16 + l15;
  const float* lbase0 = &xtile[0][((wave >> 2) * 16 + l15) * KCHUNK + hi * 2];
  const float* lbase1 = &xtile[1][((wave >> 2) * 16 + l15) * KCHUNK + hi * 2];

  // prologue: stage chunk 0 into buffer 0
  copy16_g2l(x + (size_t)gr0 * 512 + sc, &xtile[0][sr0 * KCHUNK + sc]);
  copy16_g2l(x + (size_t)gr1 * 512 + sc, &xtile[0][sr1 * KCHUNK + sc]);
  wait_async_lds();
  __syncthreads();

  v8f c = {};
  for (int ch = 0; ch < 8; ++ch) {
    if (ch < 7) {  // stage next chunk into the other buffer (no wait yet)
      const int kn = (ch + 1) * KCHUNK;
      float* dbuf = xtile[(ch + 1) & 1];
      copy16_g2l(x + (size_t)gr0 * 512 + kn + sc, &dbuf[sr0 * KCHUNK + sc]);
      copy16_g2l(x + (size_t)gr1 * 512 + kn + sc, &dbuf[sr1 * KCHUNK + sc]);
    }
    if (waveOK) {
      const float* lb = (ch & 1) ? lbase1 : lbase0;
      const int kb = ch * KCHUNK;
#pragma unroll
      for (int kk = 0; kk < KCHUNK; kk += 4) {
        v2f a = *(const v2f*)(lb + kk);                 // ds_load_b64
        v2f b;
        b.x = wc[(size_t)(kb + kk) * 64];
        b.y = wc[(size_t)(kb + kk + 1) * 64];
        c = __builtin_amdgcn_wmma_f32_16x16x4_f32(false, a, false, b, (short)0,
                                                  c, false, false);
      }
    }
    wait_async_lds();
    __syncthreads();
  }

  if (waveOK) {
    float* o = h + (size_t)(rowTile * 16 + hi * 8) * 64 + colTile * 16 + l15;
#pragma unroll
    for (int r = 0; r < 8; ++r) o[(size_t)r * 64] = c[r];
  }
}

// ---------------------------------------------------------------------------
// GEMM2: h2[N,40] = hin[N,64] @ W2[64,40]  (fp32 WMMA, 3 col tiles, pad to 48)
// Block = 192 threads = 6 waves: 2 row-tiles x 3 col-tiles.
// ---------------------------------------------------------------------------
__global__ __launch_bounds__(192) void gemm2_wmma(const float* __restrict__ hin,
                                                  const float* __restrict__ W,
                                                  float* __restrict__ h2, int n) {
  const int wave = threadIdx.x >> 5;
  const int lane = threadIdx.x & 31;
  const int rowTile = blockIdx.x * 2 + (wave >= 3 ? 1 : 0);
  if (rowTile * 16 >= n) return;
  const int colTile = wave % 3;
  const int l15 = lane & 15;
  const int hi  = lane >> 4;
  const int col = colTile * 16 + l15;            // 0..47
  const bool cok = (col < 40);
  const float* xr = hin + (size_t)(rowTile * 16 + l15) * 64 + hi * 2;
  v8f c = {};
#pragma unroll
  for (int k = 0; k < 64; k += 4) {
    v2f a, b;
    a.x = xr[k];
    a.y = xr[k + 1];
    const int kr = k + hi * 2;
    b.x = cok ? W[(size_t)kr * 40 + col] : 0.f;        // select, EXEC untouched
    b.y = cok ? W[(size_t)(kr + 1) * 40 + col] : 0.f;
    c = __builtin_amdgcn_wmma_f32_16x16x4_f32(false, a, false, b, (short)0, c,
                                              false, false);
  }
  if (cok) {
    float* o = h2 + (size_t)(rowTile * 16 + hi * 8) * 40 + col;
#pragma unroll
    for (int r = 0; r < 8; ++r) o[(size_t)r * 40] = c[r];
  }
}

// Wave per node: alpha_s[i] = h[i,:].a_s ; alpha_d[i] = h[i,:].a_d  (coalesced)
__global__ __launch_bounds__(256) void alpha_kernel(const float* __restrict__ h,
                                                    const float* __restrict__ a_s,
                                                    const float* __restrict__ a_d,
                                                    float* __restrict__ as,
                                                    float* __restrict__ ad, int n, int F) {
  int w = blockIdx.x * 8 + (threadIdx.x >> 5);
  if (w >= n) return;
  int lane = threadIdx.x & 31;
  const float* hp = h + (size_t)w * F;
  float s = 0.f, d = 0.f;
  for (int j = lane; j < F; j += 32) {
    float v = hp[j];
    s = fmaf(v, a_s[j], s);
    d = fmaf(v, a_d[j], d);
  }
  for (int o = 16; o > 0; o >>= 1) {
    s += __shfl_xor(s, o, 32);
    d += __shfl_xor(d, o, 32);
  }
  if (lane == 0) { as[w] = s; ad[w] = d; }
}

// m[i] initialized with the guaranteed self-loop score; ssum zeroed.
__global__ void init_seg_kernel(const float* __restrict__ as, const float* __restrict__ ad,
                                float* __restrict__ m, float* __restrict__ ssum, int n) {
  int i = blockIdx.x * blockDim.x + threadIdx.x;
  if (i >= n) return;
  m[i] = leakyf(as[i] + ad[i]);
  ssum[i] = 0.f;
}

__global__ void fill_kernel(float* __restrict__ p, size_t n, float v) {
  size_t i = (size_t)blockIdx.x * blockDim.x + threadIdx.x;
  if (i < n) p[i] = v;
}

__global__ void edge_max_kernel(const int* __restrict__ src, const int* __restrict__ dst,
                                const float* __restrict__ as, const float* __restrict__ ad,
                                float* __restrict__ m, int E) {
  int e = blockIdx.x * blockDim.x + threadIdx.x;
  if (e >= E) return;
  int s = src[e], d = dst[e];
  atomicMaxF(&m[d], leakyf(as[s] + ad[d]));
}

// Self-loop exp term (separate launch => non-atomic add into ssum is safe).
__global__ void self_expsum_kernel(const float* __restrict__ as, const float* __restrict__ ad,
                                   const float* __restrict__ m, float* __restrict__ exloop,
                                   float* __restrict__ ssum, int n) {
  int i = blockIdx.x * blockDim.x + threadIdx.x;
  if (i >= n) return;
  float v = __expf(leakyf(as[i] + ad[i]) - m[i]);
  exloop[i] = v;
  ssum[i] += v;
}

__global__ void edge_expsum_kernel(const int* __restrict__ src, const int* __restrict__ dst,
                                   const float* __restrict__ as, const float* __restrict__ ad,
                                   const float* __restrict__ m, float* __restrict__ ex,
                                   float* __restrict__ ssum, int E) {
  int e = blockIdx.x * blockDim.x + threadIdx.x;
  if (e >= E) return;
  int s = src[e], d = dst[e];
  float v = __expf(leakyf(as[s] + ad[d]) - m[d]);
  ex[e] = v;
  atomAddF(&ssum[d], v);
}

// One wave32 per edge: gather h[src,:], scale by attention, atomic-add to agg[dst,:].
__global__ __launch_bounds__(256) void edge_agg_kernel(const int* __restrict__ src,
                                                       const int* __restrict__ dst,
                                                       const float* __restrict__ ex,
                                                       const float* __restrict__ ssum,
                                                       const float* __restrict__ h,
                                                       float* __restrict__ agg, int E, int F) {
  int w = blockIdx.x * 8 + (threadIdx.x >> 5);
  if (w >= E) return;
  int lane = threadIdx.x & 31;
  int s = src[w], d = dst[w];
  float att = ex[w] / (ssum[d] + 1e-16f);
  const float* hs = h + (size_t)s * F;
  float* od = agg + (size_t)d * F;
  for (int j = lane; j < F; j += 32) atomAddF(&od[j], hs[j] * att);
}

// out[i,j] = elu(agg[i,j] + h1[i,j]*att_self[i] + b[j])  (in place over agg)
__global__ void finalize1_kernel(const float* __restrict__ agg, const float* __restrict__ h1,
                                 const float* __restrict__ exloop, const float* __restrict__ ssum,
                                 const float* __restrict__ b, float* __restrict__ out, int n) {
  int t = blockIdx.x * blockDim.x + threadIdx.x;
  if (t >= n * 64) return;
  int i = t >> 6, j = t & 63;
  float att = exloop[i] / (ssum[i] + 1e-16f);
  float v = agg[t] + h1[t] * att + b[j];
  out[t] = v > 0.f ? v : __expf(v) - 1.f;
}

// Per-node wave: logits = agg + h2*att_self + b2 ; then log_softmax over 40 classes.
__global__ __launch_bounds__(256) void finalize2_kernel(const float* __restrict__ agg,
                                                        const float* __restrict__ h2,
                                                        const float* __restrict__ exloop,
                                                        const float* __restrict__ ssum,
                                                        const float* __restrict__ b,
                                                        float* __restrict__ out, int n) {
  int w = blockIdx.x * 8 + (threadIdx.x >> 5);
  if (w >= n) return;
  int lane = threadIdx.x & 31;
  float att = exloop[w] / (ssum[w] + 1e-16f);
  size_t base = (size_t)w * 40;
  float v0 = agg[base + lane] + h2[base + lane] * att + b[lane];  // lane < 40 always
  float v1 = -1e30f;
  if (lane < 8) {
    int j = lane + 32;
    v1 = agg[base + j] + h2[base + j] * att + b[j];
  }
  float mx = fmaxf(v0, v1);
  for (int o = 16; o > 0; o >>= 1) mx = fmaxf(mx, __shfl_xor(mx, o, 32));
  float s = __expf(v0 - mx) + (lane < 8 ? __expf(v1 - mx) : 0.f);
  for (int o = 16; o > 0; o >>= 1) s += __shfl_xor(s, o, 32);
  float ls = __logf(s);
  out[base + lane] = v0 - mx - ls;
  if (lane < 8) out[base + lane + 32] = v1 - mx - ls;
}

// ---------------------------------------------------------------------------
extern "C" void kernel_launch(void* const* d_in, const int* in_sizes, int n_in,
                              void* d_out, int out_size, void* d_ws, size_t ws_size,
                              hipStream_t stream) {
  const float* x    = (const float*)d_in[0];
  const int*   ei   = (const int*)d_in[1];
  const float* W1   = (const float*)d_in[2];
  const float* a_s1 = (const float*)d_in[3];
  const float* a_d1 = (const float*)d_in[4];
  const float* b1   = (const float*)d_in[5];
  const float* W2   = (const float*)d_in[6];
  const float* a_s2 = (const float*)d_in[7];
  const float* a_d2 = (const float*)d_in[8];
  const float* b2   = (const float*)d_in[9];

  const int N = in_sizes[0] / 512;   // 50000
  const int E = in_sizes[1] / 2;     // 1,600,000
  const int* src = ei;
  const int* dst = ei + E;

  float* ws = (float*)d_ws;
  size_t off = 0;
  float* h1   = ws + off; off += (size_t)N * 64;
  float* agg1 = ws + off; off += (size_t)N * 64;   // becomes ELU(out1) in place
  float* h2   = ws + off; off += (size_t)N * 40;
  float* ex   = ws + off; off += (size_t)E;        // reused for both layers
  float* as   = ws + off; off += N;
  float* ad   = ws + off; off += N;
  float* m    = ws + off; off += N;
  float* ssum = ws + off; off += N;
  float* exlp = ws + off; off += N;
  float* agg2 = h1;                                // h1 dead after finalize1
  float* out  = (float*)d_out;

  const int rowTiles = (N + 15) / 16;
  const int gemmBlocks = (rowTiles + 1) / 2;
  const int nB  = (N + 255) / 256;
  const int eB  = (E + 255) / 256;
  const int neB = ((int)((size_t)N * 64) + 255) / 256;
  const int e8B = (E + 7) / 8;
  const int n8B = (N + 7) / 8;

  // ---------------- Layer 1 ----------------
  gemm1_wmma<<<gemmBlocks, 256, 0, stream>>>(x, W1, h1, N);
  alpha_kernel<<<n8B, 256, 0, stream>>>(h1, a_s1, a_d1, as, ad, N, 64);
  init_seg_kernel<<<nB, 256, 0, stream>>>(as, ad, m, ssum, N);
  fill_kernel<<<neB, 256, 0, stream>>>(agg1, (size_t)N * 64, 0.f);
  edge_max_kernel<<<eB, 256, 0, stream>>>(src, dst, as, ad, m, E);
  self_expsum_kernel<<<nB, 256, 0, stream>>>(as, ad, m, exlp, ssum, N);
  edge_expsum_kernel<<<eB, 256, 0, stream>>>(src, dst, as, ad, m, ex, ssum, E);
  edge_agg_kernel<<<e8B, 256, 0, stream>>>(src, dst, ex, ssum, h1, agg1, E, 64);
  finalize1_kernel<<<neB, 256, 0, stream>>>(agg1, h1, exlp, ssum, b1, agg1, N);

  // ---------------- Layer 2 ----------------
  gemm2_wmma<<<gemmBlocks, 192, 0, stream>>>(agg1, W2, h2, N);
  alpha_kernel<<<n8B, 256, 0, stream>>>(h2, a_s2, a_d2, as, ad, N, 40);
  init_seg_kernel<<<nB, 256, 0, stream>>>(as, ad, m, ssum, N);
  fill_kernel<<<((int)((size_t)N * 40) + 255) / 256, 256, 0, stream>>>(agg2, (size_t)N * 40, 0.f);
  edge_max_kernel<<<eB, 256, 0, stream>>>(src, dst, as, ad, m, E);
  self_expsum_kernel<<<nB, 256, 0, stream>>>(as, ad, m, exlp, ssum, N);
  edge_expsum_kernel<<<eB, 256, 0, stream>>>(src, dst, as, ad, m, ex, ssum, E);
  edge_agg_kernel<<<e8B, 256, 0, stream>>>(src, dst, ex, ssum, h2, agg2, E, 40);
  finalize2_kernel<<<n8B, 256, 0, stream>>>(agg2, h2, exlp, ssum, b2, out, N);
}